// GNN_GAT_20074677142127
// MI455X (gfx1250) — compile-verified
//
#include <hip/hip_runtime.h>
#include <math.h>

// GATv2 x3 layers for MI455X (gfx1250, wave32).
// Dense lin layers via V_WMMA_F32_16X16X4_F32 (fp32 matrix path; compute is
// not the bottleneck so full precision is free). Edge phase is L2-resident
// gather/scatter (h = 64MB < 192MB L2).

typedef __attribute__((ext_vector_type(2))) float v2f;
typedef __attribute__((ext_vector_type(8))) float v8f;

#define HEADS 5
#define CH    64
#define HC    320      // HEADS*CH
#define NDIM  128
#define EDIM  16
#define NEG_SLOPE 0.2f

// ---------------------------------------------------------------- utilities
__device__ __forceinline__ void atomicMaxF(float* addr, float val) {
    // Works for mixed-sign values when initialized to -inf.
    if (val >= 0.0f) atomicMax((int*)addr, __float_as_int(val));
    else             atomicMin((unsigned int*)addr, __float_as_uint(val));
}

__global__ void gat_fill_f32(float* __restrict__ p, float v, int n) {
    int i = blockIdx.x * blockDim.x + threadIdx.x;
    if (i < n) p[i] = v;
}

// ------------------------------------------------- self-loop attr (mean fill)
__global__ void gat_selfloop_accum(const int* __restrict__ ei, const float* __restrict__ eattr,
                                   float* __restrict__ sums, float* __restrict__ cnt, int E) {
    int i = blockIdx.x * blockDim.x + threadIdx.x;
    if (i >= E) return;
    int d = ei[E + i];  // dst row of edge_index [2,E]
#pragma unroll
    for (int k = 0; k < EDIM; ++k)
        atomicAdd(&sums[(size_t)d * EDIM + k], eattr[(size_t)i * EDIM + k]);
    atomicAdd(&cnt[d], 1.0f);
}

__global__ void gat_selfloop_final(const float* __restrict__ sums, const float* __restrict__ cnt,
                                   float* __restrict__ loop_attr, int N) {
    int i = blockIdx.x * blockDim.x + threadIdx.x;
    if (i >= N * EDIM) return;
    float c = cnt[i / EDIM];
    c = c < 1.0f ? 1.0f : c;
    loop_attr[i] = sums[i] / c;
}

// --------------------------------------------------------- WMMA fp32 GEMM
// out[N,HC] = A[N,K] @ W[K,HC] + bias ; one wave32 per 16x16 output tile.
// A layout (ISA 7.12.2, 32-bit A 16x4): lane m=lane&15, half=lane>>4 picks
// K-pair {k+2*half, k+2*half+1}.  B 4x16: lane n=lane&15, same K-pair.
// C/D 16x16: VGPR r -> row r + 8*half, col = lane&15.
__global__ void gat_gemm_wmma(const float* __restrict__ A, const float* __restrict__ W,
                              const float* __restrict__ bias, float* __restrict__ out,
                              int N, int K) {
    const int tilesCol = HC / 16;                         // 20
    int wave = (blockIdx.x * blockDim.x + threadIdx.x) >> 5;
    int lane = threadIdx.x & 31;
    int nTiles = (N / 16) * tilesCol;
    if (wave >= nTiles) return;                           // wave-uniform (grid sized exactly)

    int tr = wave / tilesCol, tc = wave % tilesCol;
    int row0 = tr * 16, col0 = tc * 16;
    int m = lane & 15, half = lane >> 4;

    const float* arow = A + (size_t)(row0 + m) * K + 2 * half;      // 8B aligned (K,2*half even)
    const float* bcol = W + (size_t)(2 * half) * HC + col0 + m;

    v8f acc = {};
    for (int k = 0; k < K; k += 4) {
        v2f a = *(const v2f*)(arow + k);                             // K rows k+2h, k+2h+1
        v2f b;
        b.x = bcol[(size_t)k * HC];
        b.y = bcol[(size_t)(k + 1) * HC];
        acc = __builtin_amdgcn_wmma_f32_16x16x4_f32(
            /*neg_a=*/false, a, /*neg_b=*/false, b,
            /*c_mod=*/(short)0, acc, /*reuse_a=*/false, /*reuse_b=*/false);
    }
    float bb = bias[col0 + m];
#pragma unroll
    for (int r = 0; r < 8; ++r)
        out[(size_t)(row0 + r + 8 * half) * HC + col0 + m] = acc[r] + bb;
}

// ------------------------------------------ edge pass A: logits + segment max
// one wave32 per edge; e = eattr @ We computed on the fly from LDS-staged We.
__global__ void gat_edge_logits(const int* __restrict__ ei, const float* __restrict__ eattr,
                                const float* __restrict__ loop_attr, const float* __restrict__ hbuf,
                                const float* __restrict__ We, const float* __restrict__ att,
                                float* __restrict__ logits, float* __restrict__ mbuf,
                                int E, int N) {
    __shared__ float sWe[EDIM * HC];
    __shared__ float sAtt[HC];
    for (int i = threadIdx.x; i < EDIM * HC; i += blockDim.x) sWe[i] = We[i];
    for (int i = threadIdx.x; i < HC; i += blockDim.x) sAtt[i] = att[i];
    __syncthreads();

    int wave = (blockIdx.x * blockDim.x + threadIdx.x) >> 5;
    int lane = threadIdx.x & 31;
    int Etot = E + N;
    if (wave >= Etot) return;

    int src, dst;
    const float* ea;
    if (wave < E) { src = ei[wave]; dst = ei[E + wave]; ea = eattr + (size_t)wave * EDIM; }
    else          { src = dst = wave - E;               ea = loop_attr + (size_t)(wave - E) * EDIM; }

    float eav[EDIM];
#pragma unroll
    for (int k = 0; k < EDIM; ++k) eav[k] = ea[k];

    const float* hs = hbuf + (size_t)src * HC;
    const float* hd = hbuf + (size_t)dst * HC;

#pragma unroll
    for (int h = 0; h < HEADS; ++h) {
        float p = 0.0f;
#pragma unroll
        for (int cc = 0; cc < 2; ++cc) {
            int idx = h * CH + cc * 32 + lane;
            float ev = 0.0f;
#pragma unroll
            for (int k = 0; k < EDIM; ++k) ev += eav[k] * sWe[k * HC + idx];
            float v = hs[idx] + hd[idx] + ev;
            v = v > 0.0f ? v : NEG_SLOPE * v;           // leaky_relu
            p += v * sAtt[idx];
        }
        for (int off = 16; off > 0; off >>= 1) p += __shfl_xor(p, off, 32);
        if (lane == 0) {
            logits[(size_t)wave * HEADS + h] = p;
            atomicMaxF(&mbuf[(size_t)dst * HEADS + h], p);
        }
    }
}

// --------------------------------------- edge pass B: exp(logit - m), denom
__global__ void gat_edge_exp(const int* __restrict__ ei, float* __restrict__ logits,
                             const float* __restrict__ mbuf, float* __restrict__ denom,
                             int E, int N) {
    int t = blockIdx.x * blockDim.x + threadIdx.x;
    int Etot = E + N;
    if (t >= Etot * HEADS) return;
    int e = t / HEADS, h = t % HEADS;
    int dst = (e < E) ? ei[E + e] : (e - E);
    float m = mbuf[(size_t)dst * HEADS + h];
    if (!(m > -INFINITY && m < INFINITY)) m = 0.0f;     // isfinite guard (matches ref)
    float ex = expf(logits[t] - m);
    logits[t] = ex;                                     // reuse buffer for ex
    atomicAdd(&denom[(size_t)dst * HEADS + h], ex);
}

// ------------------------------- edge pass C: alpha * h[src] scattered to dst
__global__ void gat_edge_scatter(const int* __restrict__ ei, const float* __restrict__ exbuf,
                                 const float* __restrict__ denom, const float* __restrict__ hbuf,
                                 float* __restrict__ accum, int E, int N) {
    int wave = (blockIdx.x * blockDim.x + threadIdx.x) >> 5;
    int lane = threadIdx.x & 31;
    int Etot = E + N;
    if (wave >= Etot) return;
    int src, dst;
    if (wave < E) { src = ei[wave]; dst = ei[E + wave]; }
    else          { src = dst = wave - E; }
    const float* hs = hbuf + (size_t)src * HC;
    float* acc = accum + (size_t)dst * HC;
#pragma unroll
    for (int h = 0; h < HEADS; ++h) {
        float alpha = exbuf[(size_t)wave * HEADS + h] /
                      (denom[(size_t)dst * HEADS + h] + 1e-16f);
#pragma unroll
        for (int cc = 0; cc < 2; ++cc) {
            int idx = h * CH + cc * 32 + lane;
            atomicAdd(&acc[idx], hs[idx] * alpha);
        }
    }
}

// -------------------------------------- node pass: head mean + bias + ELU
__global__ void gat_node_finalize(const float* __restrict__ accum, const float* __restrict__ bias,
                                  float* __restrict__ out, int N) {
    int t = blockIdx.x * blockDim.x + threadIdx.x;
    if (t >= N * CH) return;
    int n = t / CH, c = t % CH;
    float s = 0.0f;
#pragma unroll
    for (int h = 0; h < HEADS; ++h) s += accum[(size_t)n * HC + h * CH + c];
    float v = s * (1.0f / HEADS) + bias[c];
    v = v > 0.0f ? v : (expf(v) - 1.0f);                // ELU
    out[t] = v;
}

// ---------------------------------------------------------------- launcher
extern "C" void kernel_launch(void* const* d_in, const int* in_sizes, int n_in,
                              void* d_out, int out_size, void* d_ws, size_t ws_size,
                              hipStream_t stream) {
    (void)n_in; (void)out_size; (void)ws_size;
    const float* x      = (const float*)d_in[0];
    const int*   ei     = (const int*)  d_in[1];
    const float* eattr  = (const float*)d_in[2];
    const float* W0     = (const float*)d_in[3];
    const float* b0     = (const float*)d_in[4];
    const float* We0    = (const float*)d_in[5];
    const float* att0   = (const float*)d_in[6];
    const float* bias0  = (const float*)d_in[7];
    const float* W12    = (const float*)d_in[8];
    const float* b12    = (const float*)d_in[9];
    const float* We12   = (const float*)d_in[10];
    const float* att12  = (const float*)d_in[11];
    const float* bias12 = (const float*)d_in[12];

    const int N    = in_sizes[0] / NDIM;   // 50000
    const int E    = in_sizes[1] / 2;      // 400000
    const int Etot = E + N;
    float* out = (float*)d_out;

    size_t off = 0;
    auto wsalloc = [&](size_t bytes) -> float* {
        char* p = (char*)d_ws + off;
        off = (off + bytes + 255) & ~(size_t)255;
        return (float*)p;
    };
    float* hbuf   = wsalloc((size_t)N * HC * sizeof(float));       // 64 MB
    float* accum  = wsalloc((size_t)N * HC * sizeof(float));       // 64 MB
    float* act    = wsalloc((size_t)N * CH * sizeof(float));       // 12.8 MB
    float* logits = wsalloc((size_t)Etot * HEADS * sizeof(float)); // 9 MB
    float* mbuf   = wsalloc((size_t)N * HEADS * sizeof(float));
    float* denom  = wsalloc((size_t)N * HEADS * sizeof(float));
    float* lsums  = wsalloc((size_t)N * EDIM * sizeof(float));
    float* lcnt   = wsalloc((size_t)N * sizeof(float));
    float* lattr  = wsalloc((size_t)N * EDIM * sizeof(float));

    const int TB = 256;
    auto cdiv = [](int a, int b) { return (a + b - 1) / b; };

    // ---- self-loop mean attrs (once)
    gat_fill_f32<<<cdiv(N * EDIM, TB), TB, 0, stream>>>(lsums, 0.0f, N * EDIM);
    gat_fill_f32<<<cdiv(N, TB), TB, 0, stream>>>(lcnt, 0.0f, N);
    gat_selfloop_accum<<<cdiv(E, TB), TB, 0, stream>>>(ei, eattr, lsums, lcnt, E);
    gat_selfloop_final<<<cdiv(N * EDIM, TB), TB, 0, stream>>>(lsums, lcnt, lattr, N);

    // ---- 3 GATv2 layers
    for (int i = 0; i < 3; ++i) {
        const float* A    = (i == 0) ? x : act;
        const int    K    = (i == 0) ? NDIM : CH;
        const float* W    = (i == 0) ? W0    : W12    + (size_t)(i - 1) * CH * HC;
        const float* b    = (i == 0) ? b0    : b12    + (size_t)(i - 1) * HC;
        const float* We   = (i == 0) ? We0   : We12   + (size_t)(i - 1) * EDIM * HC;
        const float* att  = (i == 0) ? att0  : att12  + (size_t)(i - 1) * HC;
        const float* bias = (i == 0) ? bias0 : bias12 + (size_t)(i - 1) * CH;
        float* layerOut   = (i == 2) ? out : act;

        // h = A @ W + b   (one wave per 16x16 tile; 62500 tiles, 4 waves/block)
        int tiles = (N / 16) * (HC / 16);
        gat_gemm_wmma<<<cdiv(tiles * 32, 128), 128, 0, stream>>>(A, W, b, hbuf, N, K);

        gat_fill_f32<<<cdiv(N * HEADS, TB), TB, 0, stream>>>(mbuf, -INFINITY, N * HEADS);
        gat_fill_f32<<<cdiv(N * HEADS, TB), TB, 0, stream>>>(denom, 0.0f, N * HEADS);
        gat_fill_f32<<<cdiv(N * HC, TB), TB, 0, stream>>>(accum, 0.0f, N * HC);

        gat_edge_logits<<<cdiv(Etot * 32, TB), TB, 0, stream>>>(
            ei, eattr, lattr, hbuf, We, att, logits, mbuf, E, N);
        gat_edge_exp<<<cdiv(Etot * HEADS, TB), TB, 0, stream>>>(
            ei, logits, mbuf, denom, E, N);
        gat_edge_scatter<<<cdiv(Etot * 32, TB), TB, 0, stream>>>(
            ei, logits, denom, hbuf, accum, E, N);
        gat_node_finalize<<<cdiv(N * CH, TB), TB, 0, stream>>>(accum, bias, layerOut, N);
    }
}